// RankPooling_69827578298745
// MI455X (gfx1250) — compile-verified
//
#include <hip/hip_runtime.h>
#include <math.h>

// RankPooling via Woodbury/push-through identity:
//   w = 2C * X * (I_T + 2C * X^T X)^{-1} * y,   X:[D,T], y = 1..T
// B=256, D=768, T=128, fp32 throughout (matches reference precision).
// One workgroup (8 wave32) per batch element; Gram via V_WMMA_F32_16X16X4_F32.
// X is streamed HBM -> LDS with GLOBAL_LOAD_ASYNC_TO_LDS_B128 (ASYNCcnt),
// double-buffered so the DMA overlaps WMMA issue.

typedef float v2f __attribute__((ext_vector_type(2)));
typedef float v8f __attribute__((ext_vector_type(8)));
typedef int   v4i __attribute__((ext_vector_type(4)));
typedef __attribute__((address_space(1))) v4i gv4i;   // global int4
typedef __attribute__((address_space(3))) v4i lv4i;   // LDS int4

#define BATCH 256
#define DDIM  768
#define TDIM  128
#define TWO_C 2.0e-5f
#define CHUNK 64                 // d-rows of X staged in LDS per buffer
#define NCHUNK (DDIM / CHUNK)    // 12

#if __has_builtin(__builtin_amdgcn_global_load_async_to_lds_b128) && \
    __has_builtin(__builtin_amdgcn_s_wait_asynccnt)
#define USE_ASYNC_LDS 1
#else
#define USE_ASYNC_LDS 0
#endif

// Stage one 64x128 f32 slab (32 KB) of X into LDS. 256 threads x 8 x 16B.
__device__ __forceinline__ void stage_chunk(float* dst, const float* src, int tid) {
#if USE_ASYNC_LDS
#pragma unroll
  for (int i = 0; i < (CHUNK * TDIM / 4) / 256; ++i) {   // 8 iters
    const int idx = tid + i * 256;
    __builtin_amdgcn_global_load_async_to_lds_b128(
        (gv4i*)((v4i*)((const v4i*)src + idx)),
        (lv4i*)((v4i*)dst + idx),
        /*offset=*/0, /*cpol=*/0);
  }
#else
#pragma unroll
  for (int i = 0; i < (CHUNK * TDIM / 4) / 256; ++i) {
    const int idx = tid + i * 256;
    ((float4*)dst)[idx] = ((const float4*)src)[idx];
  }
#endif
}

__device__ __forceinline__ void wait_stage() {
#if USE_ASYNC_LDS
  __builtin_amdgcn_s_wait_asynccnt(0);
#endif
}

__global__ __launch_bounds__(256)
void rank_pool_fused(const float* __restrict__ X, float* __restrict__ W) {
  // 64KB buffer, time-multiplexed:
  //   phase 1: two 32KB X slabs (double buffer)
  //   phase 2+: M / Cholesky factor (128 x 128 f32)
  __shared__ __align__(16) float smem[TDIM * TDIM];
  __shared__ float zs[TDIM];

  const int b    = blockIdx.x;
  const int tid  = threadIdx.x;
  const int wave = tid >> 5;          // 0..7  (wave32)
  const int lane = tid & 31;
  const float* Xb = X + (size_t)b * DDIM * TDIM;

  // ---------------- Phase 1: M = X^T X via WMMA f32 16x16x4 ----------------
  // Wave w owns M rows [16w, 16w+16); 8 accumulator tiles cover all 128 cols.
  v8f acc[8] = {};

  const int colsub = lane & 15;            // 0..15 within tile
  const int ksub   = (lane >> 4) << 1;     // lanes 0-15 -> K=k0,k0+1 ; 16-31 -> k0+2,k0+3

  float* const buf0 = smem;
  float* const buf1 = smem + CHUNK * TDIM;

  stage_chunk(buf0, Xb, tid);              // preload chunk 0
  wait_stage();
  __syncthreads();

  for (int chunk = 0; chunk < NCHUNK; ++chunk) {
    const float* cur = (chunk & 1) ? buf1 : buf0;
    if (chunk + 1 < NCHUNK)                // DMA next slab while we compute
      stage_chunk((chunk & 1) ? buf0 : buf1,
                  Xb + (chunk + 1) * CHUNK * TDIM, tid);

    for (int k0 = 0; k0 < CHUNK; k0 += 4) {
      const int kk = k0 + ksub;
      // A fragment: A[t1][k] = X[k][t1]  (16x4 tile, rows = this wave's strip)
      v2f afrag;
      afrag.x = cur[kk * TDIM + wave * 16 + colsub];
      afrag.y = cur[(kk + 1) * TDIM + wave * 16 + colsub];
#pragma unroll
      for (int j = 0; j < 8; ++j) {
        // B fragment: B[k][t2] = X[k][t2]  (4x16 tile, cols = 16j..16j+15)
        v2f bfrag;
        bfrag.x = cur[kk * TDIM + j * 16 + colsub];
        bfrag.y = cur[(kk + 1) * TDIM + j * 16 + colsub];
        acc[j] = __builtin_amdgcn_wmma_f32_16x16x4_f32(
            /*neg_a=*/false, afrag, /*neg_b=*/false, bfrag,
            /*c_mod=*/(short)0, acc[j], /*reuse_a=*/false, /*reuse_b=*/false);
      }
    }
    wait_stage();        // next slab fully in LDS (own wave's issues)
    __syncthreads();     // all waves done reading `cur` + all slices visible
  }

  // Spill accumulators -> LDS as row-major M[128][128].
  // C/D layout: VGPR r -> (M = r + (lane<16 ? 0 : 8), N = lane&15).
  {
    const int m0 = wave * 16 + ((lane < 16) ? 0 : 8);
#pragma unroll
    for (int j = 0; j < 8; ++j) {
#pragma unroll
      for (int r = 0; r < 8; ++r)
        smem[(m0 + r) * TDIM + j * 16 + colsub] = acc[j][r];
    }
  }
  __syncthreads();

  // ---------------- Phase 2: A = I + 2C*M ; in-place Cholesky (lower) ------
  for (int idx = tid; idx < TDIM * TDIM; idx += 256) {
    const int r = idx >> 7, c = idx & (TDIM - 1);
    float v = TWO_C * smem[idx];
    if (r == c) v += 1.0f;
    smem[idx] = v;
  }
  __syncthreads();

  for (int k = 0; k < TDIM; ++k) {
    if (tid == 0) smem[k * TDIM + k] = sqrtf(smem[k * TDIM + k]);
    __syncthreads();
    const float inv = 1.0f / smem[k * TDIM + k];
    for (int i = k + 1 + tid; i < TDIM; i += 256)
      smem[i * TDIM + k] *= inv;
    __syncthreads();
    for (int i = k + 1 + tid; i < TDIM; i += 256) {
      const float lik = smem[i * TDIM + k];
      for (int j = k + 1; j <= i; ++j)
        smem[i * TDIM + j] -= lik * smem[j * TDIM + k];
    }
    __syncthreads();
  }

  // ---------------- Phase 3: solve L L^T z = y,  y = 1..T ------------------
  if (tid < TDIM) zs[tid] = (float)(tid + 1);
  __syncthreads();
  for (int k = 0; k < TDIM; ++k) {               // forward: L u = y
    if (tid == 0) zs[k] /= smem[k * TDIM + k];
    __syncthreads();
    if (tid > k && tid < TDIM) zs[tid] -= smem[tid * TDIM + k] * zs[k];
    __syncthreads();
  }
  for (int k = TDIM - 1; k >= 0; --k) {          // backward: L^T z = u
    if (tid == 0) zs[k] /= smem[k * TDIM + k];
    __syncthreads();
    if (tid < k) zs[tid] -= smem[k * TDIM + tid] * zs[k];
    __syncthreads();
  }

  // ---------------- Phase 4: w[d] = 2C * <X[d,:], z> -----------------------
  for (int d = tid; d < DDIM; d += 256) {
    const float4* row = (const float4*)(Xb + d * TDIM);
    float s = 0.0f;
#pragma unroll
    for (int q = 0; q < TDIM / 4; ++q) {
      const float4 v = row[q];
      s += v.x * zs[4 * q + 0] + v.y * zs[4 * q + 1] +
           v.z * zs[4 * q + 2] + v.w * zs[4 * q + 3];
    }
    W[(size_t)b * DDIM + d] = TWO_C * s;
  }
}

extern "C" void kernel_launch(void* const* d_in, const int* in_sizes, int n_in,
                              void* d_out, int out_size, void* d_ws, size_t ws_size,
                              hipStream_t stream) {
  (void)in_sizes; (void)n_in; (void)out_size; (void)d_ws; (void)ws_size;
  const float* X = (const float*)d_in[0];
  float* W = (float*)d_out;
  rank_pool_fused<<<dim3(BATCH), dim3(256), 0, stream>>>(X, W);
}